// DirAttention_80324478370322
// MI455X (gfx1250) — compile-verified
//
#include <hip/hip_runtime.h>

// MI455X / gfx1250, wave32. All contractions mapped to V_WMMA_F32_16X16X4_F32
// (exact fp32 math on the matrix pipe). Compute-bound problem (~42 GFLOP,
// ~60 MB traffic, fully L2-resident at 192 MB).
// LDS tiles are pair-interleaved along K (float2 [k/2][col]) so every WMMA
// B fragment is a single ds_load_b64 into an even-aligned VGPR pair (no
// v_mov repacking), and A-side products use packed f32 math.

typedef __attribute__((ext_vector_type(2))) float v2f;
typedef __attribute__((ext_vector_type(8))) float v8f;

#define NB 8
#define NC 64
#define NL 64
#define PLANE (NL * NL)      // 4096
#define CHW   (NC * PLANE)   // 262144

static __device__ __forceinline__ v8f wmma_f32(v2f a, v2f b, v8f c) {
  // 8-arg form: (neg_a, A, neg_b, B, c_mod, C, reuse_a, reuse_b)
  return __builtin_amdgcn_wmma_f32_16x16x4_f32(false, a, false, b, (short)0, c,
                                               false, false);
}

// ---------------------------------------------------------------------------
// Kernel 0: one-shot repack of conv weights Wo (OIHW, K-pairs 9 floats apart)
// into Wrep[(dy*3+dx)][o][gc/2] as float2 so conv A fragments are single
// global_load_b64s.   9*64*96 = 55296 float2.
// ---------------------------------------------------------------------------
__global__ __launch_bounds__(256) void wrep_kernel(const float* __restrict__ Wo,
                                                   v2f* __restrict__ Wrep) {
  const int idx = blockIdx.x * 256 + threadIdx.x;  // (dxy*64 + o)*96 + gcp
  if (idx >= 9 * 64 * 96) return;
  const int gcp = idx % 96;
  const int rest = idx / 96;
  const int o = rest % 64;
  const int dxy = rest / 64;           // dy*3+dx
  const int gc = gcp * 2;
  v2f t;
  t.x = Wo[((size_t)o * 192 + gc) * 9 + dxy];
  t.y = Wo[((size_t)o * 192 + gc + 1) * 9 + dxy];
  Wrep[idx] = t;
}

// ---------------------------------------------------------------------------
// Kernel 1: xq/xk = 1x1 conv projections.  Block = (b,h) slice, 4 waves.
// xq[b,o,h,w] = sum_c Wq[o,c] * x[b,c,h,w] + bq[o]   (same for xk)
// GEMM: A = W [o x c], B = x-slice [c x w], M=N=K=64.
// ---------------------------------------------------------------------------
__global__ __launch_bounds__(128) void qk_proj_kernel(
    const float* __restrict__ x, const float* __restrict__ Wq,
    const float* __restrict__ bq, const float* __restrict__ Wk,
    const float* __restrict__ bk, float* __restrict__ xq,
    float* __restrict__ xk) {
  __shared__ v2f xs2[32 * 65];  // c-pairs x w, padded pitch
  const int b = blockIdx.x >> 6;
  const int h = blockIdx.x & 63;
  const int tid = threadIdx.x;
  const float* xsrc = x + (size_t)b * CHW + h * NL;
  for (int idx = tid; idx < 32 * NL; idx += 128) {
    const int col = idx & 63, p = idx >> 6;
    v2f t;
    t.x = xsrc[(2 * p) * PLANE + col];
    t.y = xsrc[(2 * p + 1) * PLANE + col];
    xs2[p * 65 + col] = t;
  }
  __syncthreads();

  const int wv = tid >> 5, lane = tid & 31;
  const int lh = lane & 15;        // N (or M-row for A) within half-wave
  const int ph = lane >> 4;        // K-pair offset: lanes 16-31 use pair+1
  const int kh = ph << 1;          // K offset 0 or 2
  const int o0 = wv * 16;
  v8f aq[4], ak[4];
  for (int t = 0; t < 4; ++t) { aq[t] = 0; ak[t] = 0; }

  const v2f* Wq2 = (const v2f*)Wq;
  const v2f* Wk2 = (const v2f*)Wk;
  for (int c0 = 0; c0 < NC; c0 += 4) {
    const int ro = ((o0 + lh) * NC + c0 + kh) >> 1;
    const v2f Aq = Wq2[ro];
    const v2f Ak = Wk2[ro];
    const int bp = ((c0 >> 1) + ph) * 65;
#pragma unroll
    for (int t = 0; t < 4; ++t) {
      const v2f Bf = xs2[bp + t * 16 + lh];   // single ds_load_b64
      aq[t] = wmma_f32(Aq, Bf, aq[t]);
      ak[t] = wmma_f32(Ak, Bf, ak[t]);
    }
  }

  const int mofs = ph << 3;  // C/D: lanes 16-31 hold M+8
  for (int t = 0; t < 4; ++t) {
    const int w = t * 16 + lh;
#pragma unroll
    for (int r = 0; r < 8; ++r) {
      const int o = o0 + r + mofs;
      const size_t dst = (size_t)b * CHW + (size_t)o * PLANE + h * NL + w;
      xq[dst] = aq[t][r] + bq[o];
      xk[dst] = ak[t][r] + bk[o];
    }
  }
}

// ---------------------------------------------------------------------------
// Kernel 2: attention logits  att[b][j][o][i] = sum_k Wc[o,k] Q[b,k,i] K[b,k,j] + bc[o]
// HDIR:  k=(c,w): Q[b,c*64+w,i]=xq[b,c,i,w], K[..,j]=xk[b,c,j,w]
// !HDIR: k=(c,r): Q[b,c*64+r,i]=xq[b,c,r,i], K[..,j]=xk[b,c,r,j]
// Block = (b,j): 64x4096x64 GEMM with A formed as (Wc ⊙ kvec) packed pairs.
// ---------------------------------------------------------------------------
template <bool HDIR>
__global__ __launch_bounds__(128) void att_logits_kernel(
    const float* __restrict__ xq, const float* __restrict__ xk,
    const float* __restrict__ Wc, const float* __restrict__ bc,
    float* __restrict__ att) {
  __shared__ v2f kv2[NC * 32];    // K column, pairs along k
  __shared__ v2f qt2[32 * 65];    // Q tile  [k-pair][i], padded
  __shared__ v2f wt2[NC * 33];    // Wc tile [o][k-pair], padded
  const int b = blockIdx.x >> 6;
  const int j = blockIdx.x & 63;
  const int tid = threadIdx.x;
  const float* xqb = xq + (size_t)b * CHW;
  const float* xkb = xk + (size_t)b * CHW;

  for (int idx = tid; idx < NC * 32; idx += 128) {
    const int c = idx >> 5, p = idx & 31;
    v2f t;
    if (HDIR) {
      t = ((const v2f*)(xkb + c * PLANE + j * NL))[p];  // adjacent w
    } else {
      t.x = xkb[c * PLANE + (2 * p) * NL + j];
      t.y = xkb[c * PLANE + (2 * p + 1) * NL + j];
    }
    kv2[idx] = t;
  }

  const int wv = tid >> 5, lane = tid & 31;
  const int lh = lane & 15;
  const int ph = lane >> 4;
  const int o0 = wv * 16;
  v8f acc[4];
  for (int t = 0; t < 4; ++t) acc[t] = 0;

  for (int c = 0; c < NC; ++c) {
    if (HDIR) {  // qt2[w-pair][i] = xq[b,c,i,{2p,2p+1}] : b64 global reads
      for (int idx = tid; idx < NL * 32; idx += 128) {
        const int i = idx >> 5, p = idx & 31;
        qt2[p * 65 + i] = ((const v2f*)(xqb + c * PLANE + i * NL))[p];
      }
    } else {     // qt2[r-pair][i] from two row-contiguous reads
      for (int idx = tid; idx < 32 * NL; idx += 128) {
        const int p = idx >> 6, i = idx & 63;
        v2f t;
        t.x = xqb[c * PLANE + (2 * p) * NL + i];
        t.y = xqb[c * PLANE + (2 * p + 1) * NL + i];
        qt2[p * 65 + i] = t;
      }
    }
    for (int idx = tid; idx < NC * 32; idx += 128) {
      const int o = idx >> 5, p = idx & 31;
      wt2[o * 33 + p] = ((const v2f*)(Wc + (size_t)o * (NC * NL) + c * NL))[p];
    }
    if (c + 1 < NC) {  // global_prefetch_b8: next Wc + xq tiles
      __builtin_prefetch(Wc + (size_t)(tid >> 1) * (NC * NL) + (c + 1) * NL +
                             (tid & 1) * 32, 0, 1);
      __builtin_prefetch(xqb + (c + 1) * PLANE + tid * 32, 0, 1);
    }
    __syncthreads();

    const v2f* kvc = kv2 + c * 32;
#pragma unroll 4
    for (int k0 = 0; k0 < NL; k0 += 4) {
      const int p = (k0 >> 1) + ph;
      const v2f A = wt2[(o0 + lh) * 33 + p] * kvc[p];  // packed f32 mul
#pragma unroll
      for (int t = 0; t < 4; ++t) {
        const v2f Bf = qt2[p * 65 + t * 16 + lh];      // single ds_load_b64
        acc[t] = wmma_f32(A, Bf, acc[t]);
      }
    }
    __syncthreads();
  }

  const int mofs = ph << 3;
  float* ao = att + ((size_t)b * NL + j) * PLANE;  // [b][j][o][i]
  for (int t = 0; t < 4; ++t) {
    const int i = t * 16 + lh;
#pragma unroll
    for (int r = 0; r < 8; ++r) {
      const int o = o0 + r + mofs;
      ao[o * NL + i] = acc[t][r] + bc[o];
    }
  }
}

// ---------------------------------------------------------------------------
// Kernel 3: channel softmax of att[b][j][:, i] then out[b,c,i,j] = x * p[c].
// ---------------------------------------------------------------------------
__global__ __launch_bounds__(64) void softmax_mul_kernel(
    const float* __restrict__ att, const float* __restrict__ x,
    float* __restrict__ out) {
  const int b = blockIdx.x >> 6;
  const int j = blockIdx.x & 63;
  const int i = threadIdx.x;
  const float* ap = att + ((size_t)b * NL + j) * PLANE + i;
  float v[NC];
  float mx = -3.4e38f;
#pragma unroll
  for (int o = 0; o < NC; ++o) { v[o] = ap[o * NL]; mx = fmaxf(mx, v[o]); }
  float s = 0.f;
#pragma unroll
  for (int o = 0; o < NC; ++o) { v[o] = __expf(v[o] - mx); s += v[o]; }
  const float inv = 1.0f / s;
  const float* xb = x + (size_t)b * CHW + i * NL + j;
  float* ob = out + (size_t)b * CHW + i * NL + j;
#pragma unroll 8
  for (int c = 0; c < NC; ++c) ob[c * PLANE] = xb[c * PLANE] * v[c] * inv;
}

// ---------------------------------------------------------------------------
// Kernel 4: 3x3 conv over cat=[x, h_att, w_att] (192 ch) + bias + ReLU + BN.
// Implicit GEMM: 9 taps x (64-deep K chunks per source group).
// Block = (b,p) output row; per group stage [3 rows][32 ch-pairs][68 cols].
// ---------------------------------------------------------------------------
__global__ __launch_bounds__(256) void conv_bn_kernel(
    const float* __restrict__ x, const float* __restrict__ ha,
    const float* __restrict__ wa, const v2f* __restrict__ Wrep,
    const float* __restrict__ bo, const float* __restrict__ gamma,
    const float* __restrict__ beta, const float* __restrict__ rmean,
    const float* __restrict__ rvar, float* __restrict__ y) {
  __shared__ v2f in2[3 * 32 * 68];  // [dy][c-pair][col], 52,224 B
  const int b = blockIdx.x >> 6;
  const int p = blockIdx.x & 63;
  const int tid = threadIdx.x;
  const int wv = tid >> 5, lane = tid & 31;
  const int lh = lane & 15;
  const int ph = lane >> 4;
  const int o0 = (wv >> 1) * 16;   // 4 o-strips
  const int q0 = (wv & 1) * 32;    // 2 q-tile pairs
  v8f acc[2];
  acc[0] = 0; acc[1] = 0;

  for (int g = 0; g < 3; ++g) {
    const float* src = ((g == 0) ? x : (g == 1) ? ha : wa) + (size_t)b * CHW;
    __syncthreads();
    for (int idx = tid; idx < 3 * 32 * 68; idx += 256) {
      const int col = idx % 68;
      const int rest = idx / 68;     // dy*32 + cp
      const int dy = rest >> 5;
      const int cp = rest & 31;
      const int pp = p + dy - 1;
      const int qq = col - 1;
      v2f t; t.x = 0.f; t.y = 0.f;
      if (pp >= 0 && pp < NL && qq >= 0 && qq < NL) {
        t.x = src[(2 * cp) * PLANE + pp * NL + qq];
        t.y = src[(2 * cp + 1) * PLANE + pp * NL + qq];
      }
      in2[rest * 68 + col] = t;
    }
    __syncthreads();

    for (int dy = 0; dy < 3; ++dy) {
      for (int dx = 0; dx < 3; ++dx) {
        const size_t abase = (size_t)((dy * 3 + dx) * 64 + o0 + lh) * 96 +
                             g * 32 + ph;
#pragma unroll 4
        for (int k0 = 0; k0 < NC; k0 += 4) {
          const v2f A = Wrep[abase + (k0 >> 1)];       // single b64 load
          const int bp = (dy * 32 + (k0 >> 1) + ph) * 68;
#pragma unroll
          for (int t = 0; t < 2; ++t) {
            const v2f Bf = in2[bp + q0 + t * 16 + lh + dx];  // ds_load_b64
            acc[t] = wmma_f32(A, Bf, acc[t]);
          }
        }
      }
    }
  }

  const int mofs = ph << 3;
  for (int t = 0; t < 2; ++t) {
    const int q = q0 + t * 16 + lh;
#pragma unroll
    for (int r = 0; r < 8; ++r) {
      const int o = o0 + r + mofs;
      const float inv = gamma[o] * rsqrtf(rvar[o] + 1e-5f);
      float val = acc[t][r] + bo[o];
      val = fmaxf(val, 0.f);
      val = val * inv + (beta[o] - rmean[o] * inv);
      y[(size_t)b * CHW + (size_t)o * PLANE + p * NL + q] = val;
    }
  }
}

// ---------------------------------------------------------------------------
extern "C" void kernel_launch(void* const* d_in, const int* in_sizes, int n_in,
                              void* d_out, int out_size, void* d_ws,
                              size_t ws_size, hipStream_t stream) {
  (void)in_sizes; (void)n_in; (void)out_size; (void)ws_size;
  const float* x     = (const float*)d_in[0];
  const float* Wq    = (const float*)d_in[1];
  const float* bq    = (const float*)d_in[2];
  const float* Wk    = (const float*)d_in[3];
  const float* bk    = (const float*)d_in[4];
  const float* Wc    = (const float*)d_in[5];
  const float* bc    = (const float*)d_in[6];
  const float* Wo    = (const float*)d_in[7];
  const float* bo    = (const float*)d_in[8];
  const float* gamma = (const float*)d_in[9];
  const float* beta  = (const float*)d_in[10];
  const float* rmean = (const float*)d_in[11];
  const float* rvar  = (const float*)d_in[12];

  float* ws   = (float*)d_ws;  // 5*2,097,152 f32 + 110,592 f32 = 42.3 MB
  float* xq   = ws;
  float* xk   = ws + 1 * 2097152;
  float* attb = ws + 2 * 2097152;
  float* hatt = ws + 3 * 2097152;
  float* watt = ws + 4 * 2097152;
  v2f*   wrep = (v2f*)(ws + 5 * 2097152);
  float* y    = (float*)d_out;

  wrep_kernel<<<dim3((9 * 64 * 96 + 255) / 256), dim3(256), 0, stream>>>(Wo,
                                                                         wrep);
  qk_proj_kernel<<<dim3(NB * NL), dim3(128), 0, stream>>>(x, Wq, bq, Wk, bk,
                                                          xq, xk);
  att_logits_kernel<true><<<dim3(NB * NL), dim3(128), 0, stream>>>(xq, xk, Wc,
                                                                   bc, attb);
  softmax_mul_kernel<<<dim3(NB * NL), dim3(64), 0, stream>>>(attb, x, hatt);
  att_logits_kernel<false><<<dim3(NB * NL), dim3(128), 0, stream>>>(xq, xk, Wc,
                                                                    bc, attb);
  softmax_mul_kernel<<<dim3(NB * NL), dim3(64), 0, stream>>>(attb, x, watt);
  conv_bn_kernel<<<dim3(NB * NL), dim3(256), 0, stream>>>(
      x, hatt, watt, wrep, bo, gamma, beta, rmean, rvar, y);
}